// block_dct_callable_77996606095537
// MI455X (gfx1250) — compile-verified
//
#include <hip/hip_runtime.h>
#include <math.h>

// ---------------------------------------------------------------------------
// Batched 8x8 DCT-II via CDNA5 V_WMMA_F32_16X16X4_F32 (full fp32 matrix pipe).
// 4 blocks packed per 16x16 tile, block-diagonal DCT operators:
//   C_tile = diag(H^T,H^T) * D_tile * diag(H,H), then elementwise scale*nm.
// Roofline: 512 MB traffic @ 23.3 TB/s -> ~22 us floor (memory-bound);
// WMMA absorbs the ~95 TFLOP/s fp32 requirement off the VALU.
// Round-2: v_cos_f32 (turns) for the constant operator (kills libm cosf
// bloat), and a VGPR double-buffer prefetch of the next tile to overlap
// HBM latency with the matrix pipe.
// ---------------------------------------------------------------------------

typedef float v2f __attribute__((ext_vector_type(2)));
typedef float v4f __attribute__((ext_vector_type(4)));
typedef float v8f __attribute__((ext_vector_type(8)));

#define RSQRT2 0.70710678118654752f

__global__ __launch_bounds__(256, 2)
void dct8x8_wmma_f32(const float* __restrict__ in, float* __restrict__ out,
                     int numTiles)
{
    // Per-wave private LDS: 256 floats input tile + 256 floats staging.
    __shared__ float lds[8 * 512];
    const int lane = threadIdx.x & 31;
    const int wave = threadIdx.x >> 5;
    float* bufA = &lds[wave * 512];    // input tile, memory-linear
    float* bufB = bufA + 256;          // U staging / output staging

    const int hi = lane >> 4;          // K-half select for A/B operand layouts
    const int mn = lane & 15;          // A-operand row m == B/C-operand col n

    // Constant operator slices in A/B operand layout:
    //   lane (mn,hi), VGPR v  ->  K = 4*kk + v + 2*hi
    // A1[m][K] = same 8x8 diag block ? H[K&7][m&7] : 0   (A1 = diag(H^T,H^T))
    // A2[K][n] = same 8x8 diag block ? H[K&7][n&7] : 0   (A2 = diag(H,H))
    // Identical per-lane values under this index mapping.
    // Angle (2j+1)*k*pi/16 rad == (2j+1)*k/32 turns; v_cos_f32 takes turns.
    v2f hs[4];
    #pragma unroll
    for (int kk = 0; kk < 4; ++kk) {
        #pragma unroll
        for (int v = 0; v < 2; ++v) {
            const int K = 4 * kk + v + 2 * hi;
            const bool sameBlk = (mn < 8) == (K < 8);
            const float turns =
                (float)((2 * (K & 7) + 1) * (mn & 7)) * (1.0f / 32.0f);
            hs[kk][v] = sameBlk ? __builtin_amdgcn_cosf(turns) : 0.f;
        }
    }

    // Column normalization factor (per lane): scale * s(col), scale = 1/sqrt(16)
    const float colf = 0.25f * (((mn & 7) == 0) ? RSQRT2 : 1.f);

    const int nWaves = (int)((gridDim.x * blockDim.x) >> 5);
    int t = (int)((blockIdx.x * blockDim.x + threadIdx.x) >> 5);
    if (t >= numTiles) return;         // wave-uniform: EXEC stays all-ones

    // ---- Software pipeline: tile t in (r0,r1), tile t+nWaves prefetched ----
    const float* src = in + (size_t)t * 256 + (size_t)lane * 8;
    v4f r0 = __builtin_nontemporal_load((const v4f*)src);
    v4f r1 = __builtin_nontemporal_load((const v4f*)(src + 4));

    #pragma unroll 1
    for (;;) {
        const int  tn   = t + nWaves;
        const bool more = tn < numTiles;             // wave-uniform
        const int  tp   = more ? tn : t;             // clamped, always valid
        const float* psrc = in + (size_t)tp * 256 + (size_t)lane * 8;
        v4f p0 = __builtin_nontemporal_load((const v4f*)psrc);
        v4f p1 = __builtin_nontemporal_load((const v4f*)(psrc + 4));

        // ---- Stage current tile to LDS ----
        *(v4f*)&bufA[lane * 8]     = r0;
        *(v4f*)&bufA[lane * 8 + 4] = r1;
        asm volatile("s_wait_dscnt 0x0" ::: "memory");  // intra-wave LDS ready

        // ---- Step 1: U = diag(H^T,H^T) * D  (4x wmma f32 16x16x4) ----
        v8f U = {0.f, 0.f, 0.f, 0.f, 0.f, 0.f, 0.f, 0.f};
        #pragma unroll
        for (int kk = 0; kk < 4; ++kk) {
            v2f b;
            #pragma unroll
            for (int v = 0; v < 2; ++v) {
                const int K = 4 * kk + v + 2 * hi;
                // D[K][n] in block-packed memory layout
                b[v] = bufA[(((K >> 3) << 1) + (mn >> 3)) * 64
                            + (K & 7) * 8 + (mn & 7)];
            }
            U = __builtin_amdgcn_wmma_f32_16x16x4_f32(
                    false, hs[kk], false, b, (short)0, U, false, false);
        }

        // ---- Stage U row-major so its *columns* re-read in A layout ----
        #pragma unroll
        for (int v = 0; v < 8; ++v)
            bufB[(v + 8 * hi) * 16 + mn] = U[v];
        asm volatile("s_wait_dscnt 0x0" ::: "memory");

        // ---- Step 2: C = U * diag(H,H)  (4x wmma f32 16x16x4) ----
        v8f C = {0.f, 0.f, 0.f, 0.f, 0.f, 0.f, 0.f, 0.f};
        #pragma unroll
        for (int kk = 0; kk < 4; ++kk) {
            const int col = 4 * kk + 2 * hi;            // even -> 8B aligned
            v2f a = *(const v2f*)&bufB[mn * 16 + col];  // U[m][col..col+1]
            C = __builtin_amdgcn_wmma_f32_16x16x4_f32(
                    false, a, false, hs[kk], (short)0, C, false, false);
        }

        // ---- scale*nm, restage in output memory layout ----
        // C-layout: lane holds C[v + 8*hi][n]; row factor depends only on v.
        #pragma unroll
        for (int v = 0; v < 8; ++v) {
            const float f = colf * ((v == 0) ? RSQRT2 : 1.f);
            bufB[(2 * hi + (mn >> 3)) * 64 + v * 8 + (mn & 7)] = C[v] * f;
        }
        asm volatile("s_wait_dscnt 0x0" ::: "memory");

        // ---- Stream 1KB tile out (coalesced b128, nontemporal) ----
        float* dst = out + (size_t)t * 256 + (size_t)lane * 8;
        v4f o0 = *(const v4f*)&bufB[lane * 8];
        v4f o1 = *(const v4f*)&bufB[lane * 8 + 4];
        __builtin_nontemporal_store(o0, (v4f*)dst);
        __builtin_nontemporal_store(o1, (v4f*)(dst + 4));

        if (!more) break;
        r0 = p0; r1 = p1; t = tn;
    }
}

extern "C" void kernel_launch(void* const* d_in, const int* in_sizes, int n_in,
                              void* d_out, int out_size, void* d_ws, size_t ws_size,
                              hipStream_t stream)
{
    const float* blocks = (const float*)d_in[0];
    float* outp = (float*)d_out;

    // 4 contiguous 8x8 blocks (256 floats) per 16x16 WMMA tile
    const int numTiles = in_sizes[0] / 256;

    const int threads = 256;                       // 8 wave32 per workgroup
    int grid = 4096;                               // grid-stride over tiles
    const int wavesPerBlock = threads / 32;
    const int maxGrid = (numTiles + wavesPerBlock - 1) / wavesPerBlock;
    if (grid > maxGrid) grid = maxGrid;
    if (grid < 1) grid = 1;

    dct8x8_wmma_f32<<<grid, threads, 0, stream>>>(blocks, outp, numTiles);
}